// FAGCN_75496935129277
// MI455X (gfx1250) — compile-verified
//
#include <hip/hip_runtime.h>
#include <hip/hip_bf16.h>
#include <math.h>

typedef __attribute__((ext_vector_type(2))) float v2f;
typedef __attribute__((ext_vector_type(8))) float v8f;

#define IN_DIM  256
#define HID     64
#define OUT_DIM 40
#define EPS_F   0.3f

// ---------------- small utility kernels ----------------
__global__ void fill_zero_kernel(float* __restrict__ p, int n) {
    int i = blockIdx.x * blockDim.x + threadIdx.x;
    if (i < n) p[i] = 0.0f;
}

__global__ void degree_kernel(const int* __restrict__ dst, float* __restrict__ deg, int E) {
    int e = blockIdx.x * blockDim.x + threadIdx.x;
    if (e < E) atomicAdd(deg + dst[e], 1.0f);
}

__global__ void rsqrt_kernel(float* __restrict__ d, int n) {
    int i = blockIdx.x * blockDim.x + threadIdx.x;
    if (i < n) d[i] = rsqrtf(fmaxf(d[i], 1.0f));
}

// ---- t1 GEMM: x0 = relu(H @ W1^T + b1), fused residual init x1 = x2 = EPS*x0 ----
// One wave handles 16 rows x 64 cols (4 accumulators of 16x16), K stepped by 4,
// using V_WMMA_F32_16X16X4_F32.
__global__ void __launch_bounds__(256)
gemm_relu_wmma_kernel(const float* __restrict__ H, const float* __restrict__ W,
                      const float* __restrict__ bias,
                      float* __restrict__ X0, float* __restrict__ X1,
                      float* __restrict__ X2, int N) {
    const int lane = threadIdx.x & 31;
    const int wave = threadIdx.x >> 5;
    const int row_base = (blockIdx.x * 8 + wave) * 16;
    if (row_base >= N) return;               // wave-uniform: EXEC stays all-1s for WMMA

    const int half = lane >> 4;              // selects K pair per 32-bit A/B ISA layout
    const int lid  = lane & 15;              // M for A-frag, N for B-frag

    // A: row = row_base+lid, VGPR pair holds K = 2*half, 2*half+1 (clamped for safety)
    const int arow = (row_base + lid < N) ? (row_base + lid) : (N - 1);
    const float* aptr = H + (size_t)arow * IN_DIM + 2 * half;

    v8f acc[4];
    acc[0] = v8f{}; acc[1] = v8f{}; acc[2] = v8f{}; acc[3] = v8f{};

#pragma unroll 2
    for (int k = 0; k < IN_DIM; k += 4) {
        v2f a  = *(const v2f*)(aptr + k);
        // B tile t: B[kk][n] = W1[(16*t + n)*IN_DIM + k + kk]; lane half picks kk pair
        v2f b0 = *(const v2f*)(W + (size_t)(lid)      * IN_DIM + k + 2 * half);
        v2f b1 = *(const v2f*)(W + (size_t)(16 + lid) * IN_DIM + k + 2 * half);
        v2f b2 = *(const v2f*)(W + (size_t)(32 + lid) * IN_DIM + k + 2 * half);
        v2f b3 = *(const v2f*)(W + (size_t)(48 + lid) * IN_DIM + k + 2 * half);
        acc[0] = __builtin_amdgcn_wmma_f32_16x16x4_f32(false, a, false, b0, (short)0, acc[0], false, false);
        acc[1] = __builtin_amdgcn_wmma_f32_16x16x4_f32(false, a, false, b1, (short)0, acc[1], false, false);
        acc[2] = __builtin_amdgcn_wmma_f32_16x16x4_f32(false, a, false, b2, (short)0, acc[2], false, false);
        acc[3] = __builtin_amdgcn_wmma_f32_16x16x4_f32(false, a, false, b3, (short)0, acc[3], false, false);
    }

    // D layout: VGPR r -> row = row_base + r + 8*half, col = 16*t + lid
    if (row_base + 16 <= N) {
        // fast path (always taken when N % 16 == 0): no per-element predication
#pragma unroll
        for (int t = 0; t < 4; ++t) {
            const int col = 16 * t + lid;
            const float bc = bias[col];
#pragma unroll
            for (int r = 0; r < 8; ++r) {
                const size_t idx = (size_t)(row_base + r + 8 * half) * HID + col;
                const float v = fmaxf(acc[t][r] + bc, 0.0f);
                X0[idx] = v;
                X1[idx] = EPS_F * v;
                X2[idx] = EPS_F * v;
            }
        }
    } else {
#pragma unroll
        for (int t = 0; t < 4; ++t) {
            const int col = 16 * t + lid;
            const float bc = bias[col];
#pragma unroll
            for (int r = 0; r < 8; ++r) {
                const int row = row_base + r + 8 * half;
                if (row < N) {
                    const size_t idx = (size_t)row * HID + col;
                    const float v = fmaxf(acc[t][r] + bc, 0.0f);
                    X0[idx] = v;
                    X1[idx] = EPS_F * v;
                    X2[idx] = EPS_F * v;
                }
            }
        }
    }
}

// ---------------- FAGCN edge kernel: one wave32 per edge ----------------
// lane owns features 2*lane, 2*lane+1 (coalesced float2); gate dot reduced via shfl_xor.
__global__ void __launch_bounds__(256)
fa_edge_kernel(const float* __restrict__ x, const int* __restrict__ src,
               const int* __restrict__ dst, const float* __restrict__ d,
               const float* __restrict__ gw, const float* __restrict__ gb,
               float* __restrict__ out, int E) {
    const int lane = threadIdx.x & 31;
    const int e = (int)((blockIdx.x * (size_t)blockDim.x + threadIdx.x) >> 5);
    if (e >= E) return;

    const int s = src[e];
    const int t = dst[e];

    const float2 xs = *(const float2*)(x + (size_t)s * HID + 2 * lane);
    const float2 xt = *(const float2*)(x + (size_t)t * HID + 2 * lane);
    const float2 wd = *(const float2*)(gw + 2 * lane);        // gate_w[:HID]  (dst half)
    const float2 ws = *(const float2*)(gw + HID + 2 * lane);  // gate_w[HID:]  (src half)

    float p = xt.x * wd.x + xt.y * wd.y + xs.x * ws.x + xs.y * ws.y;
#pragma unroll
    for (int off = 16; off > 0; off >>= 1) p += __shfl_xor(p, off, 32);

    const float g    = tanhf(p + gb[0]);
    const float coef = g * d[s] * d[t];

    atomicAdd(out + (size_t)t * HID + 2 * lane,     xs.x * coef);
    atomicAdd(out + (size_t)t * HID + 2 * lane + 1, xs.y * coef);
}

// ---------------- output head: logits + row-wise log_softmax ----------------
__global__ void __launch_bounds__(256)
out_logsoftmax_kernel(const float* __restrict__ x, const float* __restrict__ W2,
                      const float* __restrict__ b2, float* __restrict__ out, int N) {
    const int i = blockIdx.x * blockDim.x + threadIdx.x;
    if (i >= N) return;

    float xi[HID];
#pragma unroll
    for (int k = 0; k < HID; ++k) xi[k] = x[(size_t)i * HID + k];

    float lg[OUT_DIM];
    float m = -INFINITY;
#pragma unroll 4
    for (int j = 0; j < OUT_DIM; ++j) {
        float s = b2[j];
#pragma unroll
        for (int k = 0; k < HID; ++k) s = fmaf(xi[k], W2[j * HID + k], s);
        lg[j] = s;
        m = fmaxf(m, s);
    }
    float se = 0.0f;
#pragma unroll 4
    for (int j = 0; j < OUT_DIM; ++j) se += expf(lg[j] - m);
    const float lse = m + logf(se);
#pragma unroll 4
    for (int j = 0; j < OUT_DIM; ++j) out[(size_t)i * OUT_DIM + j] = lg[j] - lse;
}

// ---------------- launcher ----------------
extern "C" void kernel_launch(void* const* d_in, const int* in_sizes, int n_in,
                              void* d_out, int out_size, void* d_ws, size_t ws_size,
                              hipStream_t stream) {
    const float* h    = (const float*)d_in[0];
    const int*   src  = (const int*)  d_in[1];
    const int*   dst  = (const int*)  d_in[2];
    const float* t1_w = (const float*)d_in[3];
    const float* t1_b = (const float*)d_in[4];
    const float* gw1  = (const float*)d_in[5];
    const float* gb1  = (const float*)d_in[6];
    const float* gw2  = (const float*)d_in[7];
    const float* gb2  = (const float*)d_in[8];
    const float* t2_w = (const float*)d_in[9];
    const float* t2_b = (const float*)d_in[10];

    const int N = in_sizes[0] / IN_DIM;   // 50000
    const int E = in_sizes[1];            // 800000
    const int NH = N * HID;

    // workspace layout (floats): d[N] | x0[N*HID] | x1[N*HID] | x2[N*HID]  (~38.8 MB)
    float* dnorm = (float*)d_ws;
    float* x0 = dnorm + N;
    float* x1 = x0 + (size_t)NH;
    float* x2 = x1 + (size_t)NH;

    const int T = 256;

    // degree -> d = rsqrt(max(deg,1))
    fill_zero_kernel<<<(N + T - 1) / T, T, 0, stream>>>(dnorm, N);
    degree_kernel   <<<(E + T - 1) / T, T, 0, stream>>>(dst, dnorm, E);
    rsqrt_kernel    <<<(N + T - 1) / T, T, 0, stream>>>(dnorm, N);

    // x0 = relu(h @ t1_w^T + t1_b); fused: x1 = x2 = EPS*x0  (fp32 WMMA)
    const int waves = (N + 15) / 16;
    gemm_relu_wmma_kernel<<<(waves + 7) / 8, T, 0, stream>>>(h, t1_w, t1_b, x0, x1, x2, N);

    // layer 1: x1 += scatter(x0)
    fa_edge_kernel<<<(E + 7) / 8, T, 0, stream>>>(x0, src, dst, dnorm, gw1, gb1, x1, E);

    // layer 2: x2 += scatter(x1)
    fa_edge_kernel<<<(E + 7) / 8, T, 0, stream>>>(x1, src, dst, dnorm, gw2, gb2, x2, E);

    // head: logits + log_softmax
    out_logsoftmax_kernel<<<(N + T - 1) / T, T, 0, stream>>>(x2, t2_w, t2_b, (float*)d_out, N);
}